// MatrixMPowerSeriesLayer_3925600108785
// MI455X (gfx1250) — compile-verified
//
#include <hip/hip_runtime.h>

#define NDIM 64
#define PITCH 68                       // pad rows to kill LDS bank conflicts
#define MAT_ELEMS (NDIM * PITCH)       // 4352 floats per staged matrix
#define SERIES_LEN 8
#define NBUF 8                         // Xr Xi Ar Ai Br Bi Cr Ci

typedef __attribute__((ext_vector_type(2))) float v2f;
typedef __attribute__((ext_vector_type(4))) float v4f;
typedef __attribute__((ext_vector_type(8))) float v8f;

// D(16x16) = A(16x4) * B(4x16) + C   -- fp32 WMMA, wave32
__device__ __forceinline__ v8f wmma4(v2f a, v2f b, v8f c) {
    return __builtin_amdgcn_wmma_f32_16x16x4_f32(
        /*neg_a=*/false, a, /*neg_b=*/false, b,
        /*c_mod=*/(short)0, c, /*reuse_a=*/false, /*reuse_b=*/false);
}

// Copy a row-major 64x64 fp32 matrix (global) into LDS with row pitch 68.
__device__ __forceinline__ void g2lds(float* __restrict__ dst,
                                      const float* __restrict__ src, int tid) {
#pragma unroll
    for (int j = 0; j < 4; ++j) {
        int f   = tid + 256 * j;        // float4 index 0..1023
        int row = f >> 4;               // 16 float4 per row
        int c4  = (f & 15) << 2;
        v4f v = *reinterpret_cast<const v4f*>(src + row * NDIM + c4);
        *reinterpret_cast<v4f*>(dst + row * PITCH + c4) = v;
    }
}

// A fragment: lane = M (l15), K = 2*half + v  -> contiguous pair -> one b64 load
__device__ __forceinline__ v2f loadA(const float* M, int m0, int k, int l15, int half) {
    return *reinterpret_cast<const v2f*>(M + (m0 + l15) * PITCH + k + 2 * half);
}
// B fragment: lane = N (l15), K = 2*half + v  (rows PITCH=68 floats apart -> two b32 loads)
__device__ __forceinline__ v2f loadB(const float* M, int n0, int k, int l15, int half) {
    int kk = k + 2 * half;
    v2f b;
    b.x = M[kk * PITCH + n0 + l15];
    b.y = M[(kk + 1) * PITCH + n0 + l15];
    return b;
}

__global__ void __launch_bounds__(256)
power_series_kernel(const float* __restrict__ x,
                    const float* __restrict__ coef,
                    float* __restrict__ out) {
    extern __shared__ float lds[];
    float* Xr = lds;
    float* Xi = Xr + MAT_ELEMS;
    float* Ar = Xi + MAT_ELEMS;   // power double-buffer A
    float* Ai = Ar + MAT_ELEMS;
    float* Br = Ai + MAT_ELEMS;   // power double-buffer B
    float* Bi = Br + MAT_ELEMS;
    float* Cr = Bi + MAT_ELEMS;   // staged coefficient term
    float* Ci = Cr + MAT_ELEMS;

    const int tid  = threadIdx.x;
    const int lane = tid & 31;
    const int wave = tid >> 5;
    const int l15  = lane & 15;
    const int half = lane >> 4;
    const int m0   = (wave >> 1) << 4;        // tile row
    const int n0   = (wave & 1) << 5;         // two col tiles: n0, n0+16
    const int b    = blockIdx.x;

    const float* xr = x + (size_t)b * 2 * NDIM * NDIM;
    const float* xi = xr + NDIM * NDIM;

    g2lds(Xr, xr, tid);
    g2lds(Xi, xi, tid);

    // res = I @ c[0] = c[0]  (C/D layout: VGPR v -> row m0 + v + 8*half, col n0+l15)
    v8f resr0, resi0, resr1, resi1;
    const float* cr0 = coef;
    const float* ci0 = coef + NDIM * NDIM;
#pragma unroll
    for (int v = 0; v < 8; ++v) {
        int row = m0 + v + 8 * half;
        resr0[v] = cr0[row * NDIM + n0 + l15];
        resi0[v] = ci0[row * NDIM + n0 + l15];
        resr1[v] = cr0[row * NDIM + n0 + 16 + l15];
        resi1[v] = ci0[row * NDIM + n0 + 16 + l15];
    }

    const float* Tr = Xr;   // running power T (term 1: T = X)
    const float* Ti = Xi;
    float* Ur = Ar; float* Ui = Ai;    // write target for next power
    float* Vr = Br; float* Vi = Bi;

    for (int t = 1; t < SERIES_LEN; ++t) {
        __syncthreads();   // Cr/Ci free to overwrite; X staged before first reads
        if (t + 1 < SERIES_LEN)
            __builtin_prefetch(coef + (size_t)(t + 1) * 2 * NDIM * NDIM, 0, 1);
        g2lds(Cr, coef + (size_t)t * 2 * NDIM * NDIM, tid);
        g2lds(Ci, coef + (size_t)t * 2 * NDIM * NDIM + NDIM * NDIM, tid);

        if (t >= 2) {
            // U = T * X (complex): ur = tr@xr - ti@xi ; ui = tr@xi + ti@xr
            v8f ur0 = {0,0,0,0,0,0,0,0}, ui0 = {0,0,0,0,0,0,0,0};
            v8f ur1 = {0,0,0,0,0,0,0,0}, ui1 = {0,0,0,0,0,0,0,0};
#pragma unroll
            for (int k = 0; k < NDIM; k += 4) {
                v2f arf = loadA(Tr, m0, k, l15, half);
                v2f aif = loadA(Ti, m0, k, l15, half);
                v2f nai = -aif;                       // fp32 WMMA has no A/B neg
                v2f br0 = loadB(Xr, n0,      k, l15, half);
                v2f bi0 = loadB(Xi, n0,      k, l15, half);
                v2f br1 = loadB(Xr, n0 + 16, k, l15, half);
                v2f bi1 = loadB(Xi, n0 + 16, k, l15, half);
                ur0 = wmma4(arf, br0, ur0); ur0 = wmma4(nai, bi0, ur0);
                ui0 = wmma4(arf, bi0, ui0); ui0 = wmma4(aif, br0, ui0);
                ur1 = wmma4(arf, br1, ur1); ur1 = wmma4(nai, bi1, ur1);
                ui1 = wmma4(arf, bi1, ui1); ui1 = wmma4(aif, br1, ui1);
            }
#pragma unroll
            for (int v = 0; v < 8; ++v) {
                int row = m0 + v + 8 * half;
                Ur[row * PITCH + n0 + l15]      = ur0[v];
                Ui[row * PITCH + n0 + l15]      = ui0[v];
                Ur[row * PITCH + n0 + 16 + l15] = ur1[v];
                Ui[row * PITCH + n0 + 16 + l15] = ui1[v];
            }
            Tr = Ur; Ti = Ui;
            float* s;
            s = Ur; Ur = Vr; Vr = s;
            s = Ui; Ui = Vi; Vi = s;
        }
        __syncthreads();   // new T and staged coefficients visible

        // res_r += tr@cr - ti@ci ; res_i += tr@ci (+/-) ti@cr  (reference signs)
#pragma unroll
        for (int k = 0; k < NDIM; k += 4) {
            v2f arf = loadA(Tr, m0, k, l15, half);
            v2f aif = loadA(Ti, m0, k, l15, half);
            v2f nai = -aif;
            v2f sai = (t == 1) ? aif : nai;   // '+' for first term, '-' after
            v2f br0 = loadB(Cr, n0,      k, l15, half);
            v2f bi0 = loadB(Ci, n0,      k, l15, half);
            v2f br1 = loadB(Cr, n0 + 16, k, l15, half);
            v2f bi1 = loadB(Ci, n0 + 16, k, l15, half);
            resr0 = wmma4(arf, br0, resr0); resr0 = wmma4(nai, bi0, resr0);
            resi0 = wmma4(arf, bi0, resi0); resi0 = wmma4(sai, br0, resi0);
            resr1 = wmma4(arf, br1, resr1); resr1 = wmma4(nai, bi1, resr1);
            resi1 = wmma4(arf, bi1, resi1); resi1 = wmma4(sai, br1, resi1);
        }
    }

    float* outr = out + (size_t)b * 2 * NDIM * NDIM;
    float* outi = outr + NDIM * NDIM;
#pragma unroll
    for (int v = 0; v < 8; ++v) {
        int row = m0 + v + 8 * half;
        outr[row * NDIM + n0 + l15]      = resr0[v];
        outi[row * NDIM + n0 + l15]      = resi0[v];
        outr[row * NDIM + n0 + 16 + l15] = resr1[v];
        outi[row * NDIM + n0 + 16 + l15] = resi1[v];
    }
}

extern "C" void kernel_launch(void* const* d_in, const int* in_sizes, int n_in,
                              void* d_out, int out_size, void* d_ws, size_t ws_size,
                              hipStream_t stream) {
    (void)n_in; (void)d_ws; (void)ws_size; (void)out_size;
    const float* x    = (const float*)d_in[0];
    const float* coef = (const float*)d_in[1];
    float* out        = (float*)d_out;

    const int nbatch = in_sizes[0] / (2 * NDIM * NDIM);   // 2048
    const size_t shmem = (size_t)NBUF * MAT_ELEMS * sizeof(float);  // 136 KB

    // Opt in to >64KB dynamic LDS (CDNA5 WGP has 320KB). Deterministic,
    // idempotent host call; not a stream op, safe under graph capture.
    hipFuncSetAttribute((const void*)power_series_kernel,
                        hipFuncAttributeMaxDynamicSharedMemorySize, (int)shmem);

    power_series_kernel<<<nbatch, 256, shmem, stream>>>(x, coef, out);
}